// SAGE_78417512891170
// MI455X (gfx1250) — compile-verified
//
#include <hip/hip_runtime.h>
#include <hip/hip_bf16.h>

// ---------------------------------------------------------------------------
// GraphSAGE 3-layer inference for MI455X (gfx1250, wave32).
//   layer l: out = h[:nd] @ Wself + (segsum(h[src], dst)/max(deg,1)) @ Wneigh + b
//
// Roofline: ~0.8 GB of traffic @ 23.3 TB/s (~35 us); GEMM FLOPs (~10.6 GF)
// are negligible -> fp32 end-to-end via V_WMMA_F32_16X16X4_F32.
//
// Key decisions:
//  * Layer 0 never materializes h0 = emb[input_nodes]; aggregation and the
//    GEMM self-term indirect through input_nodes (saves ~1.1 GB HBM traffic).
//  * Aggregation is CSR-based (hist -> scan -> scatter -> per-row waves)
//    instead of per-(edge,feature) float atomics: 130M global_atomic_add_f32
//    would bottleneck the L2 RMW units (~300+ us), while CSR costs only 2M
//    int atomics plus coalesced float4 row reads at full HBM bandwidth.
//  * GEMM: block = 256 threads = 8 waves = 8 M-tiles x 1 N-tile; Wself/Wneigh
//    K x 16 panels staged transposed in LDS -> inner loop is a clause of
//    global_load_b64 (A) + ds_load_2addr_b64 (B) + v_wmma_f32_16x16x4_f32.
// ---------------------------------------------------------------------------

typedef __attribute__((ext_vector_type(2))) float v2f;
typedef __attribute__((ext_vector_type(8))) float v8f;

#define IN_FEATS 128
#define N_HIDDEN 256
#define N_CLASSES 47

static const int kNDst[3]  = {67584, 6144, 1024};
static const int kNEdge[3] = {1013760, 61440, 5120};

// ---------------------------------------------------------------------------
// Phase 1: histogram of destination indices (int atomics, cheap).
// ---------------------------------------------------------------------------
__global__ __launch_bounds__(256) void sage_hist(
    const int* __restrict__ dst, int* __restrict__ cnt, int nedge)
{
    int e = blockIdx.x * 256 + threadIdx.x;
    if (e < nedge) atomicAdd(&cnt[dst[e]], 1);
}

// ---------------------------------------------------------------------------
// Phase 2: exclusive prefix sum over counts -> rowptr[n+1]. Single block of
// 1024 threads (32 waves), wave shuffle scan + LDS wave-sum scan + carry.
// ---------------------------------------------------------------------------
__global__ __launch_bounds__(1024) void sage_scan(
    const int* __restrict__ cnt, int* __restrict__ rowptr, int n)
{
    __shared__ int wsum[32];
    __shared__ int carry;
    const int tid  = threadIdx.x;
    const int lane = tid & 31;
    const int w    = tid >> 5;
    if (tid == 0) carry = 0;
    __syncthreads();

    for (int base = 0; base < n; base += 1024) {
        int i = base + tid;
        int v = (i < n) ? cnt[i] : 0;

        // inclusive scan within wave
        int x = v;
        for (int off = 1; off < 32; off <<= 1) {
            int y = __shfl_up(x, off);
            if (lane >= off) x += y;
        }
        if (lane == 31) wsum[w] = x;
        __syncthreads();

        // scan the 32 wave totals with wave 0
        if (w == 0) {
            int s = wsum[lane];
            for (int off = 1; off < 32; off <<= 1) {
                int y = __shfl_up(s, off);
                if (lane >= off) s += y;
            }
            wsum[lane] = s;
        }
        __syncthreads();

        int incl = x + (w > 0 ? wsum[w - 1] : 0) + carry;
        if (i < n) rowptr[i] = incl - v;   // exclusive
        __syncthreads();
        if (tid == 1023) carry = incl;
        __syncthreads();
    }
    if (tid == 0) rowptr[n] = carry;       // total edge count
}

// ---------------------------------------------------------------------------
// Phase 3: scatter (remapped) source indices into CSR order.
// cursor must be a copy of rowptr[0..n).
// ---------------------------------------------------------------------------
__global__ __launch_bounds__(256) void sage_scatter(
    const int* __restrict__ src, const int* __restrict__ dst,
    const int* __restrict__ remap, int* __restrict__ cursor,
    int* __restrict__ csr_src, int nedge)
{
    int e = blockIdx.x * 256 + threadIdx.x;
    if (e >= nedge) return;
    int s = src[e];
    if (remap) s = remap[s];
    int pos = atomicAdd(&cursor[dst[e]], 1);
    csr_src[pos] = s;
}

// ---------------------------------------------------------------------------
// Phase 4: per-row aggregation. One wave per dst row; lanes cover the feature
// dim with float4 (DIN=128: 1 per lane, DIN=256: 2). Coalesced 512B/1KB row
// reads, register accumulation, one plain vector store. No float atomics.
// ---------------------------------------------------------------------------
template <int DIN>
__global__ __launch_bounds__(256) void sage_row_agg(
    const float* __restrict__ h, const int* __restrict__ rowptr,
    const int* __restrict__ csr_src, float* __restrict__ agg, int ndst)
{
    const int gw   = (int)((blockIdx.x * 256 + threadIdx.x) >> 5);  // wave id
    const int lane = threadIdx.x & 31;
    if (gw >= ndst) return;

    const int beg = rowptr[gw];
    const int end = rowptr[gw + 1];
    constexpr int NV = DIN / 128;          // float4 chunks per lane

    float4 acc[NV];
#pragma unroll
    for (int v = 0; v < NV; ++v) acc[v] = make_float4(0.f, 0.f, 0.f, 0.f);

    for (int j = beg; j < end; ++j) {
        const float4* row = (const float4*)(h + (size_t)csr_src[j] * DIN);
#pragma unroll
        for (int v = 0; v < NV; ++v) {
            float4 x = row[lane + 32 * v];
            acc[v].x += x.x; acc[v].y += x.y; acc[v].z += x.z; acc[v].w += x.w;
        }
    }

    float4* out = (float4*)(agg + (size_t)gw * DIN);
#pragma unroll
    for (int v = 0; v < NV; ++v) out[lane + 32 * v] = acc[v];
}

// ---------------------------------------------------------------------------
// Fused SAGE GEMM:  Out = Hd[remap] @ Ws + (Agg/max(cnt,1)) @ Wn + Bias (+ReLU)
//
// WMMA f32 16x16x4 operand layouts (wave32):
//   A 16x4: lane L holds row m=L&15; VGPR0 = K=(L>>4)*2, VGPR1 = K+1
//   B 4x16: lane L holds col n=L&15; VGPR0 = K=(L>>4)*2, VGPR1 = K+1
//   C/D:    acc[v] = D[v + 8*(L>>4)][L&15]
//
// B panels in LDS transposed ([n][k]) so each lane's (B[ka][n],B[ka+1][n]) is
// one aligned ds_load_b64. dout=47 bounds handled in the one-time loader and
// the epilogue only; EXEC is all-1s at every WMMA; inner loop is branch-free.
// ---------------------------------------------------------------------------
template <int DIN>
__global__ __launch_bounds__(256) void sage_gemm_wmma(
    const float* __restrict__ Hd, const int* __restrict__ remap,
    const float* __restrict__ Agg, const int* __restrict__ Cnt,
    const float* __restrict__ Ws, const float* __restrict__ Wn,
    const float* __restrict__ Bias, float* __restrict__ Out,
    int dout, int relu)
{
    __shared__ float lds_s[16 * DIN];   // Wself panel, transposed [n][k]
    __shared__ float lds_n[16 * DIN];   // Wneigh panel, transposed [n][k]

    const int tid   = threadIdx.x;
    const int wave  = tid >> 5;          // 0..7 -> M tile within block
    const int lane  = tid & 31;
    const int half  = lane >> 4;         // 0/1 -> K pair within step
    const int lm    = lane & 15;         // row (A/D) or col (B/D) index
    const int ncol0 = blockIdx.y << 4;

    // ---- one-time cooperative panel load (transposed) ------------------
    for (int i = tid; i < 16 * DIN; i += 256) {
        int n = i / DIN;
        int k = i - n * DIN;
        int c = ncol0 + n;
        float vs = 0.0f, vn = 0.0f;
        if (c < dout) {
            vs = Ws[(size_t)k * dout + c];
            vn = Wn[(size_t)k * dout + c];
        }
        lds_s[i] = vs;
        lds_n[i] = vn;
    }
    __syncthreads();

    // ---- per-wave tile setup -------------------------------------------
    const int mtile = (blockIdx.x << 3) + wave;   // 8 M tiles per block
    const int mrow  = (mtile << 4) + lm;
    const int srow  = remap ? remap[mrow] : mrow;
    const int cnt   = Cnt[mrow];
    const float inv = 1.0f / (float)(cnt > 1 ? cnt : 1);

    const float* arow  = Hd  + (size_t)srow * DIN + half * 2;
    const float* grow  = Agg + (size_t)mrow * DIN + half * 2;
    const float* bsrow = lds_s + lm * DIN + half * 2;
    const float* bnrow = lds_n + lm * DIN + half * 2;

    v8f acc = {};

    // ---- self term: Hd[remap] @ Ws -------------------------------------
#pragma unroll 8
    for (int k = 0; k < DIN; k += 4) {
        v2f a = *(const v2f*)(arow + k);
        v2f b = *(const v2f*)(bsrow + k);
        acc = __builtin_amdgcn_wmma_f32_16x16x4_f32(
            false, a, false, b, (short)0, acc, false, false);
    }

    // ---- neighbor term: (Agg / max(deg,1)) @ Wn ------------------------
#pragma unroll 8
    for (int k = 0; k < DIN; k += 4) {
        v2f a = *(const v2f*)(grow + k);
        a.x *= inv;
        a.y *= inv;
        v2f b = *(const v2f*)(bnrow + k);
        acc = __builtin_amdgcn_wmma_f32_16x16x4_f32(
            false, a, false, b, (short)0, acc, false, false);
    }

    // ---- epilogue: bias, relu, bounds-checked store --------------------
    const int ncol = ncol0 + lm;
    if (ncol < dout) {
        const float bv = Bias[ncol];
        const int mbase = (mtile << 4) + (half << 3);
#pragma unroll
        for (int v = 0; v < 8; ++v) {
            float x = acc[v] + bv;
            if (relu) x = x > 0.0f ? x : 0.0f;
            Out[(size_t)(mbase + v) * dout + ncol] = x;
        }
    }
}

// ---------------------------------------------------------------------------
// Host-side orchestration (graph-capture safe: async launches, memset, d2d).
// Input order (setup_inputs dict order):
//   0 emb_table, 1 input_nodes, 2 src0, 3 dst0, 4 src1, 5 dst1, 6 src2, 7 dst2,
//   8 Wself0, 9 Wneigh0, 10 b0, 11 Wself1, 12 Wneigh1, 13 b1,
//   14 Wself2, 15 Wneigh2, 16 b2
// ---------------------------------------------------------------------------
extern "C" void kernel_launch(void* const* d_in, const int* in_sizes, int n_in,
                              void* d_out, int out_size, void* d_ws, size_t ws_size,
                              hipStream_t stream)
{
    const float* emb         = (const float*)d_in[0];
    const int*   input_nodes = (const int*)d_in[1];
    const int*   srcs[3] = {(const int*)d_in[2], (const int*)d_in[4], (const int*)d_in[6]};
    const int*   dsts[3] = {(const int*)d_in[3], (const int*)d_in[5], (const int*)d_in[7]};
    const float* Wsf[3]  = {(const float*)d_in[8],  (const float*)d_in[11], (const float*)d_in[14]};
    const float* Wnb[3]  = {(const float*)d_in[9],  (const float*)d_in[12], (const float*)d_in[15]};
    const float* Bia[3]  = {(const float*)d_in[10], (const float*)d_in[13], (const float*)d_in[16]};

    // Workspace carve-out (~125 MB total).
    char* ws = (char*)d_ws;
    size_t off = 0;
    auto take = [&](size_t bytes) -> void* {
        void* p = ws + off;
        off = (off + bytes + 255) & ~(size_t)255;
        return p;
    };
    float* h1   = (float*)take((size_t)kNDst[0] * N_HIDDEN * sizeof(float)); // layer0 out
    float* h2   = (float*)take((size_t)kNDst[1] * N_HIDDEN * sizeof(float)); // layer1 out
    float* agg0 = (float*)take((size_t)kNDst[0] * IN_FEATS * sizeof(float));
    float* agg1 = (float*)take((size_t)kNDst[1] * N_HIDDEN * sizeof(float));
    float* agg2 = (float*)take((size_t)kNDst[2] * N_HIDDEN * sizeof(float));
    // CSR scratch (sized for the largest layer, reused sequentially).
    int* cnt     = (int*)take((size_t)kNDst[0] * sizeof(int));
    int* rowptr  = (int*)take(((size_t)kNDst[0] + 1) * sizeof(int));
    int* cursor  = (int*)take((size_t)kNDst[0] * sizeof(int));
    int* csr_src = (int*)take((size_t)kNEdge[0] * sizeof(int));

    const float* hin[3]   = {emb, h1, h2};           // layer-0 input = raw table
    const int*   remap[3] = {input_nodes, nullptr, nullptr};
    float*       aggs[3]  = {agg0, agg1, agg2};
    float*       hout[3]  = {h1, h2, (float*)d_out};
    const int    dins[3]  = {IN_FEATS, N_HIDDEN, N_HIDDEN};
    const int    douts[3] = {N_HIDDEN, N_HIDDEN, N_CLASSES};

    for (int l = 0; l < 3; ++l) {
        const int nd = kNDst[l];
        const int ne = kNEdge[l];
        const int eblocks = (ne + 255) / 256;

        // --- build CSR: hist -> scan -> scatter -------------------------
        hipMemsetAsync(cnt, 0, (size_t)nd * sizeof(int), stream);
        sage_hist<<<eblocks, 256, 0, stream>>>(dsts[l], cnt, ne);
        sage_scan<<<1, 1024, 0, stream>>>(cnt, rowptr, nd);
        hipMemcpyAsync(cursor, rowptr, (size_t)nd * sizeof(int),
                       hipMemcpyDeviceToDevice, stream);
        sage_scatter<<<eblocks, 256, 0, stream>>>(
            srcs[l], dsts[l], remap[l], cursor, csr_src, ne);

        // --- aggregate: one wave per dst row ----------------------------
        if (dins[l] == IN_FEATS)
            sage_row_agg<IN_FEATS><<<nd / 8, 256, 0, stream>>>(
                hin[l], rowptr, csr_src, aggs[l], nd);
        else
            sage_row_agg<N_HIDDEN><<<nd / 8, 256, 0, stream>>>(
                hin[l], rowptr, csr_src, aggs[l], nd);

        // --- fused dual-GEMM + bias + relu ------------------------------
        dim3 grid(nd / 128, (douts[l] + 15) / 16);   // 8 M-tiles per block
        if (dins[l] == IN_FEATS)
            sage_gemm_wmma<IN_FEATS><<<grid, 256, 0, stream>>>(
                hin[l], remap[l], aggs[l], cnt,
                Wsf[l], Wnb[l], Bia[l], hout[l],
                douts[l], (l != 2) ? 1 : 0);
        else
            sage_gemm_wmma<N_HIDDEN><<<grid, 256, 0, stream>>>(
                hin[l], remap[l], aggs[l], cnt,
                Wsf[l], Wnb[l], Bia[l], hout[l],
                douts[l], (l != 2) ? 1 : 0);
    }
}